// GSGNet_25606595019232
// MI455X (gfx1250) — compile-verified
//
#include <hip/hip_runtime.h>
#include <math.h>

#define NN 200000
#define NE 6400000
#define FI 26
#define FH 40
#define FO 24
#define KP1 52             // layer-1 GEMM K: FI(self) + FI(mean), exact multiple of 4
#define KP2 40             // layer-2 GEMM K
#define NP 48              // padded N (3 x 16 tiles)
#define SPW 48             // sp row width: [s2(24) | p2(24)]
#define ROWT (NN / 16)     // 12500 row tiles, exact
#define WPB 8              // waves per block

typedef __attribute__((ext_vector_type(2))) float v2f;
typedef __attribute__((ext_vector_type(8))) float v8f;

// ---------------- utility kernels ----------------

__global__ void k_zero(float* __restrict__ p, int n) {
  int i = blockIdx.x * blockDim.x + threadIdx.x;
  if (i < n) p[i] = 0.0f;
}

__global__ void k_deg(const int* __restrict__ dst, float* __restrict__ deg) {
  int e = blockIdx.x * blockDim.x + threadIdx.x;
  if (e < NE) atomicAdd(&deg[dst[e]], 1.0f);
}

__global__ void k_deginv(float* __restrict__ deg) {
  int i = blockIdx.x * blockDim.x + threadIdx.x;
  if (i < NN) {
    float d = deg[i];
    deg[i] = (d > 0.0f) ? (1.0f / d) : 0.0f;
  }
}

// one feature per lane: a wave handles one edge's whole feature row (L2-resident gather)
__global__ void k_scat1(const float* __restrict__ feat, const int* __restrict__ src,
                        const int* __restrict__ dst, float* __restrict__ agg) {
  int t = blockIdx.x * blockDim.x + threadIdx.x;
  int e = t >> 5, f = t & 31;
  if (e >= NE || f >= FI) return;
  atomicAdd(&agg[dst[e] * FI + f], feat[src[e] * FI + f]);
}

// scatter the *projected* layer-2 neighbor features (24 atomics/edge instead of 40)
__global__ void k_scat2(const float* __restrict__ sp, const int* __restrict__ src,
                        const int* __restrict__ dst, float* __restrict__ agg2) {
  int t = blockIdx.x * blockDim.x + threadIdx.x;
  int e = t >> 5, f = t & 31;
  if (e >= NE || f >= FO) return;
  atomicAdd(&agg2[dst[e] * FO + f], sp[src[e] * SPW + FO + f]);
}

// ---------------- WMMA GEMM kernels (LDS-staged, branch-free inner loop) ----------------

// h1 = relu([feat | mean] @ [W1_self; W1_neigh] + b1)
__global__ __launch_bounds__(256) void k_gemm1(
    const float* __restrict__ feat, const float* __restrict__ agg,
    const float* __restrict__ dinv, const float* __restrict__ Ws,
    const float* __restrict__ Wn, const float* __restrict__ bias,
    float* __restrict__ h1) {
  __shared__ float wlds[NP * KP1];          // col-major, zero-padded: wlds[n*KP1 + k]
  __shared__ float alds[WPB * 16 * KP1];    // per-wave [16][KP1] = [feat | mean] tile

  int tid = threadIdx.x;
  int wv = tid >> 5, lane = tid & 31;
  int wid = blockIdx.x * WPB + wv;
  bool act = (wid < ROWT);

  // stage weights once per block (branches happen here, not in the hot loop)
  for (int idx = tid; idx < NP * KP1; idx += 256) {
    int n = idx / KP1, k = idx - n * KP1;
    float v = 0.0f;
    if (n < FH) v = (k < FI) ? Ws[k * FH + n] : Wn[(k - FI) * FH + n];
    wlds[idx] = v;
  }

  // stage this wave's A tile: rows = 16 nodes, cols = [feat(26) | mean(26)], coalesced
  float* amy = alds + wv * (16 * KP1);
  if (act) {
    int base = wid * 16;
    for (int idx = lane; idx < 16 * FI; idx += 32) {
      int r = idx / FI, c = idx - r * FI;
      amy[r * KP1 + c] = feat[(base + r) * FI + c];
    }
    for (int idx = lane; idx < 16 * FI; idx += 32) {
      int r = idx / FI, c = idx - r * FI;
      amy[r * KP1 + FI + c] = agg[(base + r) * FI + c] * dinv[base + r];
    }
  }
  __syncthreads();
  if (!act) return;

  int nl = lane & 15;                       // M for A frag, N for B frag
  int kb = (lane >> 4) << 1;                // K sub-offset per lane half
  const float* arow = amy + nl * KP1;

  v8f c0 = {0, 0, 0, 0, 0, 0, 0, 0};
  v8f c1 = {0, 0, 0, 0, 0, 0, 0, 0};
  v8f c2 = {0, 0, 0, 0, 0, 0, 0, 0};

#pragma unroll
  for (int k0 = 0; k0 < KP1; k0 += 4) {
    int k = k0 + kb;
    v2f a   = *(const v2f*)(arow + k);                   // ds_load_b64
    v2f b0  = *(const v2f*)(wlds + nl * KP1 + k);        // {W[k][n], W[k+1][n]}
    v2f b1v = *(const v2f*)(wlds + (16 + nl) * KP1 + k);
    v2f b2v = *(const v2f*)(wlds + (32 + nl) * KP1 + k);
    c0 = __builtin_amdgcn_wmma_f32_16x16x4_f32(false, a, false, b0,  (short)0, c0, false, false);
    c1 = __builtin_amdgcn_wmma_f32_16x16x4_f32(false, a, false, b1v, (short)0, c1, false, false);
    c2 = __builtin_amdgcn_wmma_f32_16x16x4_f32(false, a, false, b2v, (short)0, c2, false, false);
  }

  int mb = wid * 16 + ((lane >> 4) << 3);   // C layout: vgpr j -> rows j / j+8
#pragma unroll
  for (int j = 0; j < 8; ++j) {
    float* hr = h1 + (mb + j) * FH;
    hr[nl]      = fmaxf(c0[j] + bias[nl], 0.0f);
    hr[16 + nl] = fmaxf(c1[j] + bias[16 + nl], 0.0f);
    if (32 + nl < FH) hr[32 + nl] = fmaxf(c2[j] + bias[32 + nl], 0.0f);
  }
}

// sp = h1 @ [W2_self | W2_neigh]   (no bias; b2 applied in epilogue kernel)
__global__ __launch_bounds__(256) void k_gemm2(
    const float* __restrict__ h1, const float* __restrict__ Ws,
    const float* __restrict__ Wn, float* __restrict__ sp) {
  __shared__ float wlds[NP * KP2];          // col-major: wlds[n*KP2 + k]
  __shared__ float alds[WPB * 16 * KP2];

  int tid = threadIdx.x;
  int wv = tid >> 5, lane = tid & 31;
  int wid = blockIdx.x * WPB + wv;
  bool act = (wid < ROWT);

  for (int idx = tid; idx < NP * KP2; idx += 256) {
    int n = idx / KP2, k = idx - n * KP2;
    wlds[idx] = (n < FO) ? Ws[k * FO + n] : Wn[k * FO + (n - FO)];
  }

  float* amy = alds + wv * (16 * KP2);
  if (act) {
    int base = wid * 16;
    for (int idx = lane; idx < 16 * KP2; idx += 32)
      amy[idx] = h1[base * KP2 + idx];      // fully contiguous copy
  }
  __syncthreads();
  if (!act) return;

  int nl = lane & 15;
  int kb = (lane >> 4) << 1;
  const float* arow = amy + nl * KP2;

  v8f c0 = {0, 0, 0, 0, 0, 0, 0, 0};
  v8f c1 = {0, 0, 0, 0, 0, 0, 0, 0};
  v8f c2 = {0, 0, 0, 0, 0, 0, 0, 0};

#pragma unroll
  for (int k0 = 0; k0 < KP2; k0 += 4) {
    int k = k0 + kb;
    v2f a   = *(const v2f*)(arow + k);
    v2f b0  = *(const v2f*)(wlds + nl * KP2 + k);
    v2f b1v = *(const v2f*)(wlds + (16 + nl) * KP2 + k);
    v2f b2v = *(const v2f*)(wlds + (32 + nl) * KP2 + k);
    c0 = __builtin_amdgcn_wmma_f32_16x16x4_f32(false, a, false, b0,  (short)0, c0, false, false);
    c1 = __builtin_amdgcn_wmma_f32_16x16x4_f32(false, a, false, b1v, (short)0, c1, false, false);
    c2 = __builtin_amdgcn_wmma_f32_16x16x4_f32(false, a, false, b2v, (short)0, c2, false, false);
  }

  int mb = wid * 16 + ((lane >> 4) << 3);
#pragma unroll
  for (int j = 0; j < 8; ++j) {
    float* sr = sp + (mb + j) * SPW;
    sr[nl]      = c0[j];
    sr[16 + nl] = c1[j];
    sr[32 + nl] = c2[j];
  }
}

// out = log_softmax(s2 + deg_inv * agg2 + b2)
__global__ void k_final(const float* __restrict__ sp, const float* __restrict__ agg2,
                        const float* __restrict__ dinv, const float* __restrict__ b2,
                        float* __restrict__ out) {
  int i = blockIdx.x * blockDim.x + threadIdx.x;
  if (i >= NN) return;
  float di = dinv[i];
  float v[FO];
  float mx = -1e30f;
#pragma unroll
  for (int c = 0; c < FO; ++c) {
    float val = sp[i * SPW + c] + di * agg2[i * FO + c] + b2[c];
    v[c] = val;
    mx = fmaxf(mx, val);
  }
  float s = 0.0f;
#pragma unroll
  for (int c = 0; c < FO; ++c) s += expf(v[c] - mx);
  float l = logf(s);
#pragma unroll
  for (int c = 0; c < FO; ++c) out[i * FO + c] = v[c] - mx - l;
}

// ---------------- launcher ----------------

extern "C" void kernel_launch(void* const* d_in, const int* in_sizes, int n_in,
                              void* d_out, int out_size, void* d_ws, size_t ws_size,
                              hipStream_t stream) {
  (void)in_sizes; (void)n_in; (void)out_size; (void)ws_size;
  const float* feat = (const float*)d_in[0];
  const int*   src  = (const int*)d_in[1];
  const int*   dst  = (const int*)d_in[2];
  const float* W1s  = (const float*)d_in[3];
  const float* W1n  = (const float*)d_in[4];
  const float* b1   = (const float*)d_in[5];
  const float* W2s  = (const float*)d_in[6];
  const float* W2n  = (const float*)d_in[7];
  const float* b2   = (const float*)d_in[8];
  float* out = (float*)d_out;

  // workspace layout (floats): dinv[NN] | aggbuf[NN*26] | h1[NN*40] | sp[NN*48]  = 92 MB
  float* dinv   = (float*)d_ws;
  float* aggbuf = dinv + NN;
  float* h1     = aggbuf + (size_t)NN * FI;
  float* sp     = h1 + (size_t)NN * FH;

  const int tb = 256;
  const int edge_lane_blocks = (NE * 32 + tb - 1) / tb;   // one feature per lane
  const int gemm_blocks = (ROWT + WPB - 1) / WPB;         // one 16-row tile per wave

  // degree + layer-1 aggregate accumulator zeroing (contiguous 27*NN floats)
  k_zero<<<(27 * NN + tb - 1) / tb, tb, 0, stream>>>(dinv, 27 * NN);
  k_deg<<<(NE + tb - 1) / tb, tb, 0, stream>>>(dst, dinv);
  k_deginv<<<(NN + tb - 1) / tb, tb, 0, stream>>>(dinv);

  // layer 1: aggregate raw features, then fused [feat|mean] GEMM + bias + relu
  k_scat1<<<edge_lane_blocks, tb, 0, stream>>>(feat, src, dst, aggbuf);
  k_gemm1<<<gemm_blocks, tb, 0, stream>>>(feat, aggbuf, dinv, W1s, W1n, b1, h1);

  // layer 2: project first (24-wide), scatter projected, then epilogue softmax
  k_zero<<<(FO * NN + tb - 1) / tb, tb, 0, stream>>>(aggbuf, FO * NN);
  k_gemm2<<<gemm_blocks, tb, 0, stream>>>(h1, W2s, W2n, sp);
  k_scat2<<<edge_lane_blocks, tb, 0, stream>>>(sp, src, dst, aggbuf);
  k_final<<<(NN + tb - 1) / tb, tb, 0, stream>>>(sp, aggbuf, dinv, b2, out);
}